// Attention_71940702208453
// MI455X (gfx1250) — compile-verified
//
#include <hip/hip_runtime.h>
#include <math.h>

typedef __attribute__((ext_vector_type(2))) float v2f;
typedef __attribute__((ext_vector_type(8))) float v8f;

namespace cfg {
constexpr int B = 32, N = 577, C = 1024, H = 16, D = 64;
constexpr int M1 = B * N;                      // 18464 token rows
constexpr int SLD = 592;                       // 577 padded to 16-multiple
constexpr float SC = 0.125f;                   // HEAD_DIM^-0.5
// float offsets inside workspace
constexpr long QKV_OFF = 0;
constexpr long QKV_SZ  = (long)M1 * 3 * C;     // 56,721,408
constexpr long S_OFF   = QKV_OFF + QKV_SZ;     // stage1 scores [512,64,592]; reused for stage2 [512,577,64]
constexpr long S_SZ    = 512L * 64 * SLD;      // 19,398,656
constexpr long QP_OFF  = S_OFF + S_SZ;         // pooled queries [B,64,C]
constexpr long QP_SZ   = (long)B * 64 * C;
constexpr long QD_OFF  = QP_OFF + QP_SZ;       // Q_delta [B,H,64,64]
constexpr long QD_SZ   = 512L * 64 * 64;
constexpr long AO_OFF  = QD_OFF + QD_SZ;       // pre-projection attention out [B,N,C]
}

// ---------------------------------------------------------------------------
// Batched NT GEMM on V_WMMA_F32_16X16X4_F32:
//   C[m,n] = sum_k A[m,k] * B[n,k] (+ bias[n])
//   A elem:  A + m*lda + k           (k contiguous -> float2 loads)
//   B elem:  B + n*ldbN + k*ldbK     (BK1: ldbK==1 -> float2 loads)
// Wave tile 32x32 (2x2 WMMA frags); 8 waves arranged WM x WN per block.
// Batch z = blockIdx.z decomposed as (b = z/H, h = z%H) with per-b/per-h strides.
// ---------------------------------------------------------------------------
template<int WM, int WN, bool BK1, bool HAS_BIAS>
__global__ __launch_bounds__(256) void gemm_nt_wmma(
    const float* __restrict__ A, const float* __restrict__ Bm,
    const float* __restrict__ bias, float* __restrict__ Cp,
    int M, int Nn, int K,
    long lda, long ldbN, long ldbK, long ldc,
    long sAb, long sAh, long sBb, long sBh, long sCb, long sCh, int Hdiv)
{
  const int z = blockIdx.z;
  const int b = z / Hdiv, h = z % Hdiv;
  A  += (long)b * sAb + (long)h * sAh;
  Bm += (long)b * sBb + (long)h * sBh;
  Cp += (long)b * sCb + (long)h * sCh;

  const int lane = threadIdx.x & 31;
  const int wave = threadIdx.x >> 5;       // 0..7
  const int wm   = wave / WN;              // 0..WM-1
  const int wn   = wave % WN;              // 0..WN-1
  const int half = lane >> 4;              // K-pair selector
  const int l16  = lane & 15;              // M row (A) / N col (B)

  const int m0 = blockIdx.x * (WM * 32) + wm * 32;
  const int n0 = blockIdx.y * (WN * 32) + wn * 32;

  int ar[2], br[2];
#pragma unroll
  for (int i = 0; i < 2; ++i) {
    int r = m0 + i * 16 + l16; ar[i] = (r < M)  ? r : (M - 1);
    int c = n0 + i * 16 + l16; br[i] = (c < Nn) ? c : (Nn - 1);
  }

  v8f acc[2][2] = {};
  for (int k = 0; k < K; k += 4) {
    const int kk = k + 2 * half;
    v2f af[2], bf[2];
#pragma unroll
    for (int i = 0; i < 2; ++i)
      af[i] = *(const v2f*)(A + (long)ar[i] * lda + kk);
#pragma unroll
    for (int i = 0; i < 2; ++i) {
      if (BK1) {
        bf[i] = *(const v2f*)(Bm + (long)br[i] * ldbN + kk);
      } else {
        const float* p = Bm + (long)br[i] * ldbN + (long)kk * ldbK;
        bf[i].x = p[0];
        bf[i].y = p[ldbK];
      }
    }
#pragma unroll
    for (int i = 0; i < 2; ++i)
#pragma unroll
      for (int j = 0; j < 2; ++j)
        acc[i][j] = __builtin_amdgcn_wmma_f32_16x16x4_f32(
            false, af[i], false, bf[j], (short)0, acc[i][j], false, false);
  }

#pragma unroll
  for (int i = 0; i < 2; ++i)
#pragma unroll
    for (int j = 0; j < 2; ++j)
#pragma unroll
      for (int r = 0; r < 8; ++r) {
        const int row = m0 + i * 16 + r + 8 * half;
        const int col = n0 + j * 16 + l16;
        if (row < M && col < Nn) {
          float v = acc[i][j][r];
          if constexpr (HAS_BIAS) v += bias[col];
          Cp[(long)row * ldc + col] = v;
        }
      }
}

// ---------------------------------------------------------------------------
// Exact 3x3 average pool of the 24x24 non-cls query tokens -> Qp [B,64,C]
// ---------------------------------------------------------------------------
__global__ __launch_bounds__(256) void pool_q_kernel(
    const float* __restrict__ qkv, float* __restrict__ Qp)
{
  using namespace cfg;
  const int idx = blockIdx.x * 256 + threadIdx.x;
  if (idx >= B * 64 * C) return;
  const int c = idx % C;
  const int p = (idx / C) & 63;
  const int b = idx / (C * 64);
  const int py = p >> 3, px = p & 7;
  const float* base = qkv + (long)b * N * 3 * C + c;   // Xq slice (first C cols)
  float s = 0.f;
#pragma unroll
  for (int dy = 0; dy < 3; ++dy)
#pragma unroll
    for (int dx = 0; dx < 3; ++dx) {
      const int tok = (3 * py + dy) * 24 + (3 * px + dx);
      s += base[(long)tok * 3 * C];
    }
  Qp[idx] = s * (1.0f / 9.0f);
}

// ---------------------------------------------------------------------------
// Row softmax (in place): p[i] = softmax(scale * p[i]) over i<n; zero padding
// up to ldr so padded K-columns contribute 0 in the following GEMM.
// ---------------------------------------------------------------------------
__global__ __launch_bounds__(256) void softmax_rows_kernel(
    float* __restrict__ S, int n, int ldr, float scale)
{
  float* p = S + (long)blockIdx.x * ldr;
  __shared__ float red[256];
  const int t = threadIdx.x;

  float mx = -INFINITY;
  for (int i = t; i < n; i += 256) mx = fmaxf(mx, p[i] * scale);
  red[t] = mx; __syncthreads();
  for (int s = 128; s > 0; s >>= 1) {
    if (t < s) red[t] = fmaxf(red[t], red[t + s]);
    __syncthreads();
  }
  mx = red[0]; __syncthreads();

  float sum = 0.f;
  for (int i = t; i < n; i += 256) {
    const float e = expf(p[i] * scale - mx);
    p[i] = e; sum += e;
  }
  red[t] = sum; __syncthreads();
  for (int s = 128; s > 0; s >>= 1) {
    if (t < s) red[t] += red[t + s];
    __syncthreads();
  }
  const float inv = 1.0f / red[0];
  for (int i = t; i < n; i += 256) p[i] *= inv;
  for (int i = n + t; i < ldr; i += 256) p[i] = 0.0f;
}

// ---------------------------------------------------------------------------
extern "C" void kernel_launch(void* const* d_in, const int* in_sizes, int n_in,
                              void* d_out, int out_size, void* d_ws, size_t ws_size,
                              hipStream_t stream)
{
  using namespace cfg;
  const float* X     = (const float*)d_in[0];
  const float* Wqkv  = (const float*)d_in[1];
  const float* Wproj = (const float*)d_in[2];
  const float* bproj = (const float*)d_in[3];
  float* out = (float*)d_out;
  float* ws  = (float*)d_ws;

  float* qkv = ws + QKV_OFF;   // [B*N, 3C]
  float* S   = ws + S_OFF;     // scores (stage1 then stage2)
  float* Qp  = ws + QP_OFF;    // [B,64,C]
  float* Qd  = ws + QD_OFF;    // [B,H,64,64]
  float* AO  = ws + AO_OFF;    // [B,N,C]

  // 1) qkv = X @ Wqkv^T          M=18464, N=3072, K=1024
  gemm_nt_wmma<2, 4, true, false><<<dim3(289, 24, 1), 256, 0, stream>>>(
      X, Wqkv, nullptr, qkv, M1, 3 * C, C,
      C, C, 1, 3 * C, 0, 0, 0, 0, 0, 0, 1);

  // 2) pooled queries
  pool_q_kernel<<<(B * 64 * C + 255) / 256, 256, 0, stream>>>(qkv, Qp);

  // 3) S1[z,q,k] = Qp_h . K_h     per (b,h): M=64, N=577, K=64
  gemm_nt_wmma<2, 4, true, false><<<dim3(1, 5, 512), 256, 0, stream>>>(
      Qp, qkv + C, nullptr, S, 64, N, D,
      C, 3L * C, 1, SLD,
      64L * C, D, (long)N * 3 * C, D, 64L * SLD * H, 64L * SLD, H);

  // 4) softmax over keys (scale applied here), zero 577..591 padding
  softmax_rows_kernel<<<512 * 64, 256, 0, stream>>>(S, N, SLD, SC);

  // 5) Qd = A1 @ V_h              per (b,h): M=64, N=64(d), K=592 (pad rows have A1=0)
  gemm_nt_wmma<2, 4, false, false><<<dim3(1, 1, 512), 256, 0, stream>>>(
      S, qkv + 2 * C, nullptr, Qd, 64, D, SLD,
      SLD, 1, 3L * C, D,
      64L * SLD * H, 64L * SLD, (long)N * 3 * C, D, 4096L * H, 4096L, H);

  // 6) S2[z,n,q] = Qh . Qp_h      per (b,h): M=577, N=64, K=64 (reuse S buffer)
  gemm_nt_wmma<4, 2, true, false><<<dim3(5, 1, 512), 256, 0, stream>>>(
      qkv, Qp, nullptr, S, N, 64, D,
      3L * C, C, 1, 64,
      (long)N * 3 * C, D, 64L * C, D, 577L * 64 * H, 577L * 64, H);

  // 7) softmax over the 64 pooled queries
  softmax_rows_kernel<<<512 * N, 256, 0, stream>>>(S, 64, 64, SC);

  // 8) AO[b,n,h*64+d] = A2 @ Qd   per (b,h): M=577, N=64(d), K=64
  gemm_nt_wmma<4, 2, false, false><<<dim3(5, 1, 512), 256, 0, stream>>>(
      S, Qd, nullptr, AO, N, D, 64,
      64, 1, 64, C,
      577L * 64 * H, 577L * 64, 4096L * H, 4096L, 577L * C, (long)D, H);

  // 9) out = AO @ Wproj^T + b     M=18464, N=1024, K=1024
  gemm_nt_wmma<2, 4, true, true><<<dim3(289, 8, 1), 256, 0, stream>>>(
      AO, Wproj, bproj, out, M1, C, C,
      C, C, 1, C, 0, 0, 0, 0, 0, 0, 1);
}